// RTEIntegrationLayer_52510270161122
// MI455X (gfx1250) — compile-verified
//
#include <hip/hip_runtime.h>
#include <math.h>
#include <stdint.h>

// ---------------- problem / tiling constants ----------------
#define R_POINTS      32
#define ROW_STRIDE_B  144                       // 128B row + 16B pad (16B aligned, low LDS bank conflict)
#define ARR_B         (R_POINTS * ROW_STRIDE_B) // 4608 B : one array (32 rows) in LDS
#define BUF_B         (2 * ARR_B)               // 9216 B : fv + T for one tile
#define WAVE_B        (2 * BUF_B)               // 18432 B: double buffered per wave
#define WAVES_PER_BLOCK 2
#define BLOCK_THREADS (WAVES_PER_BLOCK * 32)
#define LDS_B         (WAVES_PER_BLOCK * WAVE_B) // 36864 B per block

// physical constants (double-evaluated, folded to f32)
#define PI_D 3.14159265358979323846
__device__ constexpr float KC  = (float)(6.0 * PI_D * 0.37 / 6.33e-07); // kappa coeff
__device__ constexpr float SCc = (float)(5.67e-08 / PI_D);              // sigma/pi

// ---------------- CDNA5 async global -> LDS (ASYNCcnt path) ----------------
__device__ __forceinline__ void async_b128(unsigned ldsAddr, const void* sbase, unsigned voff) {
    // GVS mode: 64-bit SGPR base + 32-bit per-lane VGPR offset; VDST = LDS byte address
    asm volatile("global_load_async_to_lds_b128 %0, %1, %2"
                 :: "v"(ldsAddr), "v"(voff), "s"(sbase)
                 : "memory");
}

// issue one 32-row tile (fv then T) = 16 async b128 instructions, each 512B contiguous
__device__ __forceinline__ void issue_tile(unsigned bufAbs, const float* Tf, const float* fvf,
                                           int tile, int sub, int phase) {
    const float* fG = fvf + (size_t)tile * (32 * R_POINTS);
    const float* tG = Tf  + (size_t)tile * (32 * R_POINTS);
#pragma unroll
    for (int i = 0; i < 8; ++i) {
        int ri = i * 4 + sub;                       // row index this lane's 16B chunk belongs to
        async_b128(bufAbs + (unsigned)(ri * ROW_STRIDE_B + phase), fG,
                   (unsigned)(ri * 128 + phase));
    }
#pragma unroll
    for (int i = 0; i < 8; ++i) {
        int ri = i * 4 + sub;
        async_b128(bufAbs + (unsigned)(ARR_B + ri * ROW_STRIDE_B + phase), tG,
                   (unsigned)(ri * 128 + phase));
    }
}

// ---------------- per-row RTE math (all in registers, backward suffix scan) ----------------
__device__ __forceinline__ float rte_math(const float* Tt, const float* kap, const float* rg) {
    float S = 0.0f;
    float t2  = Tt[31] * Tt[31];
    float acc = (rg[31] - rg[30]) * (SCc * kap[31] * (t2 * t2));   // tau[31] = 0
#pragma unroll
    for (int i = 30; i >= 0; --i) {
        // S = suffix trapezoid of kappa; tau[i] = -S  (sign per reference's flipped dx)
        S = fmaf(0.5f * (rg[i + 1] - rg[i]), kap[i] + kap[i + 1], S);
        float a2 = Tt[i] * Tt[i];
        float gi = SCc * kap[i] * (a2 * a2) * __expf(S);           // exp(-tau) = exp(S)
        float w  = rg[i + 1] - rg[(i == 0) ? 0 : (i - 1)];         // trapezoid weights (x2*0.5 folded)
        acc = fmaf(w, gi, acc);
    }
    return acc;
}

__device__ __forceinline__ float row_from_global(const float* Trow, const float* fvRow,
                                                 const float* rg) {
    float kap[32], Tt[32];
#pragma unroll
    for (int g = 0; g < 8; ++g) {
        float4 f = *reinterpret_cast<const float4*>(fvRow + g * 4);
        float4 t = *reinterpret_cast<const float4*>(Trow + g * 4);
        kap[4*g+0] = KC * f.x; kap[4*g+1] = KC * f.y; kap[4*g+2] = KC * f.z; kap[4*g+3] = KC * f.w;
        Tt [4*g+0] = t.x;      Tt [4*g+1] = t.y;      Tt [4*g+2] = t.z;      Tt [4*g+3] = t.w;
    }
    return rte_math(Tt, kap, rg);
}

// ---------------- kernel ----------------
extern "C" __global__ void __launch_bounds__(BLOCK_THREADS)
rte_async_kernel(const float* __restrict__ Tf, const float* __restrict__ fvf,
                 const float* __restrict__ rgp, float* __restrict__ out,
                 int rows, int fullTiles, int numWaves)
{
    __shared__ __align__(16) float smemf[LDS_B / 4];

    const int lane  = (int)(threadIdx.x & 31u);
    const int wib   = (int)(threadIdx.x >> 5);
    const int gwave = (int)blockIdx.x * WAVES_PER_BLOCK + wib;

    // uniform radial grid -> scalar regs (hoisted out of the tile loop)
    float rg[R_POINTS];
#pragma unroll
    for (int i = 0; i < R_POINTS; ++i) rg[i] = rgp[i];

    // absolute LDS byte address of this wave's region (flat-shared low 32 bits == LDS offset)
    const unsigned smemByte = (unsigned)(uintptr_t)(&smemf[0]);
    const unsigned waveAbs  = smemByte + (unsigned)(wib * WAVE_B);
    const unsigned waveRel  = (unsigned)(wib * WAVE_B);

    const int sub   = lane >> 3;          // which row group within a 512B async chunk
    const int phase = (lane & 7) * 16;    // 16B sub-chunk within a row

    int tile = gwave;
    if (tile < fullTiles) issue_tile(waveAbs, Tf, fvf, tile, sub, phase);   // prime buffer 0

    int buf = 0;
    for (; tile < fullTiles; tile += numWaves) {
        int next = tile + numWaves;
        if (next < fullTiles) {
            issue_tile(waveAbs + (unsigned)((buf ^ 1) * BUF_B), Tf, fvf, next, sub, phase);
            asm volatile("s_wait_asynccnt 0x10" ::: "memory");  // current tile's 16 loads done
        } else {
            asm volatile("s_wait_asynccnt 0x0" ::: "memory");
        }

        // lane owns row `lane` of the tile: read 2x 128B from LDS (b128, padded stride)
        unsigned rel    = waveRel + (unsigned)(buf * BUF_B);
        unsigned fvWord = (rel + (unsigned)(lane * ROW_STRIDE_B)) >> 2;
        unsigned tWord  = (rel + (unsigned)(ARR_B + lane * ROW_STRIDE_B)) >> 2;

        float kap[32], Tt[32];
#pragma unroll
        for (int g = 0; g < 8; ++g) {
            float4 f = *reinterpret_cast<const float4*>(&smemf[fvWord + g * 4]);
            float4 t = *reinterpret_cast<const float4*>(&smemf[tWord  + g * 4]);
            kap[4*g+0] = KC * f.x; kap[4*g+1] = KC * f.y; kap[4*g+2] = KC * f.z; kap[4*g+3] = KC * f.w;
            Tt [4*g+0] = t.x;      Tt [4*g+1] = t.y;      Tt [4*g+2] = t.z;      Tt [4*g+3] = t.w;
        }
        out[(size_t)tile * 32 + lane] = rte_math(Tt, kap, rg);   // coalesced 128B store
        buf ^= 1;
    }

    // tail rows (rows % 32): direct global path, handled by wave 0
    int tailStart = fullTiles * 32;
    if (gwave == 0 && tailStart + lane < rows) {
        int row = tailStart + lane;
        out[row] = row_from_global(Tf + (size_t)row * R_POINTS,
                                   fvf + (size_t)row * R_POINTS, rg);
    }
}

// ---------------- launch ----------------
extern "C" void kernel_launch(void* const* d_in, const int* in_sizes, int n_in,
                              void* d_out, int out_size, void* d_ws, size_t ws_size,
                              hipStream_t stream) {
    (void)n_in; (void)d_ws; (void)ws_size; (void)out_size;
    const float* Tf  = (const float*)d_in[0];
    const float* fvf = (const float*)d_in[1];
    const float* rgp = (const float*)d_in[2];
    float* out = (float*)d_out;

    int rows      = in_sizes[0] / R_POINTS;
    int fullTiles = rows / 32;

    long long needBlocks = ((long long)fullTiles + WAVES_PER_BLOCK - 1) / WAVES_PER_BLOCK;
    int grid = (int)(needBlocks < 4096 ? needBlocks : 4096);   // grid-stride; 1 tile/wave until the 4096-block cap
    if (grid < 1) grid = 1;
    int numWaves = grid * WAVES_PER_BLOCK;

    rte_async_kernel<<<grid, BLOCK_THREADS, 0, stream>>>(Tf, fvf, rgp, out,
                                                         rows, fullTiles, numWaves);
}